// SpectralConv1dODE_5669356830904
// MI455X (gfx1250) — compile-verified
//
#include <hip/hip_runtime.h>

// MI455X / gfx1250 spectral-ODE implementation.
// Matrix math: V_WMMA_F32_16X16X4_F32 (fp32-exact).
// W(m) LDS staging in the RK4 kernel: Tensor Data Mover (TENSOR_LOAD_TO_LDS)
// with hardware row-padding (128 DWORDs + 2 pad -> 130-float rows).

typedef __attribute__((ext_vector_type(2))) float v2f;
typedef __attribute__((ext_vector_type(8))) float v8f;
typedef __attribute__((ext_vector_type(4))) unsigned int v4u;
typedef __attribute__((ext_vector_type(8))) int v8i;
typedef __attribute__((ext_vector_type(4))) int v4i;

#define WMMA4(a, b, c) \
  __builtin_amdgcn_wmma_f32_16x16x4_f32(false, (a), false, (b), (short)0, (c), false, false)

#if __has_builtin(__builtin_amdgcn_tensor_load_to_lds) && \
    __has_builtin(__builtin_amdgcn_s_wait_tensorcnt)
#define USE_TDM 1
#else
#define USE_TDM 0
#endif

constexpr int B_ = 32;    // batch
constexpr int N_ = 512;   // nodes
constexpr int C_ = 128;   // channels (Cin == Cout)
constexpr int M_ = 64;    // retained Fourier modes
constexpr int T_ = 20;    // time points
constexpr int PAD = 130;  // LDS row pad (floats) -> bank-conflict free
constexpr float PI2 = 6.28318530717958647692f;

// ---------------------------------------------------------------- tables ----
__global__ void k_tables(float* __restrict__ Cf, float* __restrict__ Sf,
                         float* __restrict__ G) {
  int gsz = gridDim.x * blockDim.x;
  int gid = blockIdx.x * blockDim.x + threadIdx.x;
  for (int i = gid; i < M_ * N_; i += gsz) {
    int m = i >> 9, n = i & (N_ - 1);
    int r = (m * n) & (N_ - 1);                 // exact range reduction
    float th = (PI2 / (float)N_) * (float)r;
    Cf[i] = cosf(th);
    Sf[i] = -sinf(th);
  }
  for (int i = gid; i < N_ * C_; i += gsz) {
    int n = i >> 7, k = i & (C_ - 1), m = k >> 1;
    int r = (m * n) & (N_ - 1);
    float th = (PI2 / (float)N_) * (float)r;
    float c = ((m == 0) ? 1.0f : 2.0f) * (1.0f / (float)N_);
    G[i] = (k & 1) ? (-c * sinf(th)) : (c * cosf(th));
  }
}

// ------------------------------------------------------------- W repack ----
// weights: [Cin][Cout][M][2] -> Wtr/Wti: [m][o][i]  (contiguous in i)
__global__ void k_repackw(const float* __restrict__ w, float* __restrict__ wtr,
                          float* __restrict__ wti) {
  int gid = blockIdx.x * blockDim.x + threadIdx.x;
  if (gid >= M_ * C_ * C_) return;
  int i = gid & (C_ - 1);
  int o = (gid >> 7) & (C_ - 1);
  int m = gid >> 14;
  int src = ((i * C_ + o) * M_ + m) * 2;
  wtr[gid] = w[src];
  wti[gid] = w[src + 1];
}

// ------------------------------------------------------ stage 1: fwd DFT ----
__global__ void __launch_bounds__(256) k_dft(const float* __restrict__ x,
                                             const float* __restrict__ Cf,
                                             const float* __restrict__ Sf,
                                             float* __restrict__ yre,
                                             float* __restrict__ yim) {
  int b = blockIdx.x;
  int w = threadIdx.x >> 5, lane = threadIdx.x & 31;
  int lo = lane & 15, hi = lane >> 4;
  int i0 = w * 16 + lo;                         // this wave's Cin column tile
  v8f z = {0.f, 0.f, 0.f, 0.f, 0.f, 0.f, 0.f, 0.f};
  v8f aR[4] = {z, z, z, z};
  v8f aI[4] = {z, z, z, z};
#pragma unroll 4
  for (int s = 0; s < N_ / 4; ++s) {
    int kb = 4 * s + 2 * hi;
    v2f bf;
    bf.x = x[(b * N_ + kb) * C_ + i0];
    bf.y = x[(b * N_ + kb + 1) * C_ + i0];
#pragma unroll
    for (int mt = 0; mt < 4; ++mt) {
      int row = mt * 16 + lo;
      v2f ca = *reinterpret_cast<const v2f*>(Cf + row * N_ + kb);
      v2f sa = *reinterpret_cast<const v2f*>(Sf + row * N_ + kb);
      aR[mt] = WMMA4(ca, bf, aR[mt]);
      aI[mt] = WMMA4(sa, bf, aI[mt]);
    }
  }
#pragma unroll
  for (int mt = 0; mt < 4; ++mt)
#pragma unroll
    for (int v = 0; v < 8; ++v) {
      int m = mt * 16 + hi * 8 + v;
      yre[(m * B_ + b) * C_ + i0] = aR[mt][v];
      yim[(m * B_ + b) * C_ + i0] = aI[mt][v];
    }
}

// ------------------------------------------------------ TDM descriptor ------
#if USE_TDM
// Issue a TDM load of a 128x128 f32 tile at `gaddr` into LDS byte offset
// `ldsOff`, padding each 128-DWORD row by 2 DWORDs (130-float LDS rows).
__device__ __forceinline__ void tdm_load_w_tile(unsigned long long gaddr,
                                                unsigned int ldsOff) {
  v4u g0;
  g0[0] = 1u;                                   // count=1, user mode
  g0[1] = ldsOff;                               // lds_addr (bytes)
  g0[2] = (unsigned int)(gaddr & 0xffffffffull);
  g0[3] = (unsigned int)(gaddr >> 32) | (2u << 30);   // addr[56:32] | type=2
  v8i g1;
  // data_size=2 (4B) | pad_enable | pad_interval=6 (128 dw) | pad_amount=1 (2 dw)
  g1[0] = (int)((2u << 16) | (1u << 20) | (6u << 22) | (1u << 25));
  g1[1] = (int)((unsigned)(C_ & 0xffff) << 16);       // tensor_dim0[15:0] @ bits 63:48
  g1[2] = (int)((unsigned)(C_ & 0xffff) << 16);       // dim0 hi=0 | tensor_dim1[15:0]
  g1[3] = (int)((unsigned)(C_ & 0xffff) << 16);       // dim1 hi=0 | tile_dim0=128
  g1[4] = (int)(C_ & 0xffff);                         // tile_dim1=128 | tile_dim2=0
  g1[5] = (int)C_;                                    // tensor_dim0_stride[31:0]=128
  g1[6] = 0;                                          // stride0 hi | dim1_stride lo
  g1[7] = 0;                                          // dim1_stride hi
  v4i z4 = {0, 0, 0, 0};
  v8i z8 = {0, 0, 0, 0, 0, 0, 0, 0};
  // 6-arg form (clang-23 / therock-10.0): (g0, g1, g2, g3, extra, cpol)
  __builtin_amdgcn_tensor_load_to_lds(g0, g1, z4, z4, z8, 0);
}
#endif

// ------------------------------------------------------ stage 2: RK4 step ---
// One WG per mode m. W(m) + y + u staged in LDS, RK4 accumulators in VGPRs.
// f(y)[b,o] = sum_i y[b,i] * W[i,o]   (complex; Re = P - Q, Im = R).
__global__ void __launch_bounds__(256) k_rk4(
    const float* __restrict__ ycr, const float* __restrict__ yci,
    const float* __restrict__ wtr, const float* __restrict__ wti,
    float* __restrict__ ynr, float* __restrict__ yni, float dt) {
  extern __shared__ float sm[];
  float* Wr = sm;                    // [128][PAD]  W^T real (o-major, i inner)
  float* Wi = Wr + C_ * PAD;         // [128][PAD]  W^T imag
  float* Yr = Wi + C_ * PAD;         // [32][PAD]   y_t real
  float* Yi = Yr + B_ * PAD;
  float* Ur = Yi + B_ * PAD;         // [32][PAD]   current f input
  float* Ui = Ur + B_ * PAD;
  int m = blockIdx.x;
  int tid = threadIdx.x;

#if USE_TDM
  if ((tid >> 5) == 0) {             // one wave issues both tile DMAs
    unsigned long long wb = (unsigned long long)(uintptr_t)wtr +
                            (unsigned long long)m * (C_ * C_ * 4ull);
    unsigned long long wbi = (unsigned long long)(uintptr_t)wti +
                             (unsigned long long)m * (C_ * C_ * 4ull);
    tdm_load_w_tile(wb, 0u);
    tdm_load_w_tile(wbi, (unsigned int)(C_ * PAD * 4));
  }
#else
  for (int idx = tid; idx < C_ * C_; idx += 256) {
    int o = idx >> 7, i = idx & (C_ - 1);
    Wr[o * PAD + i] = wtr[m * C_ * C_ + idx];
    Wi[o * PAD + i] = wti[m * C_ * C_ + idx];
  }
#endif
  for (int idx = tid; idx < B_ * C_; idx += 256) {
    int b = idx >> 7, c = idx & (C_ - 1);
    float vr = ycr[m * B_ * C_ + idx];
    float vi = yci[m * B_ * C_ + idx];
    Yr[b * PAD + c] = vr; Ur[b * PAD + c] = vr;
    Yi[b * PAD + c] = vi; Ui[b * PAD + c] = vi;
  }
#if USE_TDM
  if ((tid >> 5) == 0) __builtin_amdgcn_s_wait_tensorcnt(0);
#endif
  __syncthreads();

  int w = tid >> 5, lane = tid & 31, lo = lane & 15, hi = lane >> 4;
  int o0 = w * 16;                   // this wave's Cout tile (shared by 2 b-tiles)
  v8f z = {0.f, 0.f, 0.f, 0.f, 0.f, 0.f, 0.f, 0.f};
  v8f accR[2] = {z, z}, accI[2] = {z, z};
  const float wk[4]  = {dt / 6.f, dt / 3.f, dt / 3.f, dt / 6.f};
  const float cin[4] = {dt * 0.5f, dt * 0.5f, dt, 0.f};

#pragma unroll
  for (int st = 0; st < 4; ++st) {
    v8f P[2] = {z, z}, Q[2] = {z, z}, R[2] = {z, z};
#pragma unroll 4
    for (int s = 0; s < C_ / 4; ++s) {
      int kb = 4 * s + 2 * hi;
      v2f br = *reinterpret_cast<const v2f*>(Wr + (o0 + lo) * PAD + kb);
      v2f bi = *reinterpret_cast<const v2f*>(Wi + (o0 + lo) * PAD + kb);
#pragma unroll
      for (int bt = 0; bt < 2; ++bt) {
        v2f ar = *reinterpret_cast<const v2f*>(Ur + (bt * 16 + lo) * PAD + kb);
        v2f ai = *reinterpret_cast<const v2f*>(Ui + (bt * 16 + lo) * PAD + kb);
        P[bt] = WMMA4(ar, br, P[bt]);   // Ar*Br
        Q[bt] = WMMA4(ai, bi, Q[bt]);   // Ai*Bi  (Re = P - Q)
        R[bt] = WMMA4(ar, bi, R[bt]);   // Ar*Bi
        R[bt] = WMMA4(ai, br, R[bt]);   // + Ai*Br (Im)
      }
    }
    __syncthreads();                    // all reads of U done before rewrite
#pragma unroll
    for (int bt = 0; bt < 2; ++bt)
#pragma unroll
      for (int v = 0; v < 8; ++v) {
        float kre = P[bt][v] - Q[bt][v];
        float kim = R[bt][v];
        accR[bt][v] += wk[st] * kre;
        accI[bt][v] += wk[st] * kim;
        if (st < 3) {
          int rr = bt * 16 + hi * 8 + v, cc = o0 + lo;
          Ur[rr * PAD + cc] = Yr[rr * PAD + cc] + cin[st] * kre;
          Ui[rr * PAD + cc] = Yi[rr * PAD + cc] + cin[st] * kim;
        }
      }
    __syncthreads();
  }
#pragma unroll
  for (int bt = 0; bt < 2; ++bt)
#pragma unroll
    for (int v = 0; v < 8; ++v) {
      int rr = bt * 16 + hi * 8 + v, cc = o0 + lo;
      ynr[(m * B_ + rr) * C_ + cc] = Yr[rr * PAD + cc] + accR[bt][v];
      yni[(m * B_ + rr) * C_ + cc] = Yi[rr * PAD + cc] + accI[bt][v];
    }
}

// ------------------------------------------------- stage 3: truncated irfft --
__global__ void __launch_bounds__(256) k_irfft(const float* __restrict__ yre,
                                               const float* __restrict__ yim,
                                               const float* __restrict__ G,
                                               float* __restrict__ out, int t) {
  extern __shared__ float Ht[];       // [128 o][PAD] with k inner (Ht[o][k])
  int b = blockIdx.x, tid = threadIdx.x;
  for (int idx = tid; idx < C_ * M_; idx += 256) {
    int o = idx & (C_ - 1), mm = idx >> 7;
    Ht[o * PAD + 2 * mm]     = yre[(mm * B_ + b) * C_ + o];
    Ht[o * PAD + 2 * mm + 1] = yim[(mm * B_ + b) * C_ + o];
  }
  __syncthreads();
  int w = tid >> 5, lane = tid & 31, lo = lane & 15, hi = lane >> 4;
  int o0 = w * 16;
  v2f bf[32];                          // hoist all B fragments (K=128)
#pragma unroll
  for (int s = 0; s < 32; ++s)
    bf[s] = *reinterpret_cast<const v2f*>(Ht + (o0 + lo) * PAD + 4 * s + 2 * hi);
  v8f z = {0.f, 0.f, 0.f, 0.f, 0.f, 0.f, 0.f, 0.f};
  for (int nt = 0; nt < N_ / 16; ++nt) {
    v8f D = z;
#pragma unroll 8
    for (int s = 0; s < 32; ++s) {
      v2f a = *reinterpret_cast<const v2f*>(G + (nt * 16 + lo) * C_ + 4 * s + 2 * hi);
      D = WMMA4(a, bf[s], D);
    }
    int base = (b * T_ + t) * N_ + nt * 16 + hi * 8;
#pragma unroll
    for (int v = 0; v < 8; ++v)
      out[(base + v) * C_ + o0 + lo] = D[v];
  }
}

// ----------------------------------------------------------------- launch ----
extern "C" void kernel_launch(void* const* d_in, const int* in_sizes, int n_in,
                              void* d_out, int out_size, void* d_ws, size_t ws_size,
                              hipStream_t stream) {
  (void)in_sizes; (void)n_in; (void)out_size; (void)ws_size;
  const float* x   = (const float*)d_in[0];
  const float* wts = (const float*)d_in[2];   // d_in[1] = t grid (uniform, dt=1/19)
  float* out = (float*)d_out;
  float* ws  = (float*)d_ws;

  size_t off = 0;
  float* Cf  = ws + off; off += (size_t)M_ * N_;
  float* Sf  = ws + off; off += (size_t)M_ * N_;
  float* G   = ws + off; off += (size_t)N_ * C_;
  float* Wtr = ws + off; off += (size_t)M_ * C_ * C_;
  float* Wti = ws + off; off += (size_t)M_ * C_ * C_;
  float* yAr = ws + off; off += (size_t)M_ * B_ * C_;
  float* yAi = ws + off; off += (size_t)M_ * B_ * C_;
  float* yBr = ws + off; off += (size_t)M_ * B_ * C_;
  float* yBi = ws + off; off += (size_t)M_ * B_ * C_;   // ~13.1 MB total

  const float dt = 1.0f / 19.0f;
  const size_t rk4_lds = (size_t)(2 * C_ * PAD + 4 * B_ * PAD) * sizeof(float);
  const size_t ir_lds  = (size_t)(C_ * PAD) * sizeof(float);
  (void)hipFuncSetAttribute((const void*)k_rk4,
                            hipFuncAttributeMaxDynamicSharedMemorySize, (int)rk4_lds);
  (void)hipFuncSetAttribute((const void*)k_irfft,
                            hipFuncAttributeMaxDynamicSharedMemorySize, (int)ir_lds);

  k_tables<<<256, 256, 0, stream>>>(Cf, Sf, G);
  k_repackw<<<(M_ * C_ * C_ + 255) / 256, 256, 0, stream>>>(wts, Wtr, Wti);
  k_dft<<<B_, 256, 0, stream>>>(x, Cf, Sf, yAr, yAi);

  float *cr = yAr, *ci = yAi, *nr = yBr, *ni = yBi;
  for (int t = 0; t < T_; ++t) {
    k_irfft<<<B_, 256, ir_lds, stream>>>(cr, ci, G, out, t);
    if (t < T_ - 1) {
      k_rk4<<<M_, 256, rk4_lds, stream>>>(cr, ci, Wtr, Wti, nr, ni, dt);
      float* tp;
      tp = cr; cr = nr; nr = tp;
      tp = ci; ci = ni; ni = tp;
    }
  }
}